// S4DKernel_51951924412416
// MI455X (gfx1250) — compile-verified
//
#include <hip/hip_runtime.h>
#include <math.h>

// S4D kernel generation for MI455X (gfx1250, wave32).
// K[h,l] = 2*Re( sum_n Ct[h,n] * exp(dtA[h,n]*l) ),  H=1024, N/2=32, L=4096.
// Decompose l = 64*l1 + l0 -> exact per-head 64x64x64 f32 GEMM on the WMMA pipe
// (V_WMMA_F32_16X16X4_F32), slashing transcendental work 32x.

typedef float v2f __attribute__((ext_vector_type(2)));
typedef float v8f __attribute__((ext_vector_type(8)));

#define HN        32      // N/2 complex modes
#define LTILE     64      // l0 dimension (inner)
#define L1DIM     64      // l1 dimension (outer), L = 64*64
#define SMS       65      // LDS row stride (pad: 64 banks -> conflict-free)

__global__ __launch_bounds__(128) void s4d_vandermonde_wmma(
    const float* __restrict__ C_real,      // (H, 32, 2)
    const float* __restrict__ log_dt,      // (H,)
    const float* __restrict__ log_A_real,  // (H, 32)
    const float* __restrict__ A_imag,      // (H, 32)
    float* __restrict__ K_out)             // (H, 4096)
{
    __shared__ float smA[L1DIM][SMS];      // Aa[l1][k] : [2Wr | -2Wi]
    __shared__ float smB[2 * HN][SMS];     // Bb[k][l0] : [Qr ; Qi]
    __shared__ float s_dre[HN], s_dim[HN], s_ctr[HN], s_cti[HN];

    const int h    = blockIdx.x;
    const int t    = threadIdx.x;          // 0..127
    const int lane = t & 31;
    const int wave = t >> 5;               // 0..3

    // ---- Step 1: per-(h,n) scalars: dtA and Ct = C*(exp(dtA)-1)/A ----------
    if (t < HN) {
        const int   n   = t;
        const float dt  = expf(log_dt[h]);
        const float Are = -expf(log_A_real[h * HN + n]);
        const float Aim = A_imag[h * HN + n];
        const float dre = Are * dt;
        const float dim = Aim * dt;
        const float ex  = expf(dre);
        const float Ere = ex * cosf(dim) - 1.0f;   // exp(dtA)-1
        const float Eim = ex * sinf(dim);
        const float inv = 1.0f / (Are * Are + Aim * Aim);
        const float Br  = (Ere * Are + Eim * Aim) * inv;  // (exp(dtA)-1)/A
        const float Bi  = (Eim * Are - Ere * Aim) * inv;
        const float Cr  = C_real[(h * HN + n) * 2 + 0];
        const float Ci  = C_real[(h * HN + n) * 2 + 1];
        s_dre[n] = dre;
        s_dim[n] = dim;
        s_ctr[n] = Cr * Br - Ci * Bi;
        s_cti[n] = Cr * Bi + Ci * Br;
    }
    __syncthreads();

    // ---- Step 2: Bb[k][l0] = [Qr ; Qi], Q = exp(dtA * l0) ------------------
    for (int idx = t; idx < HN * LTILE; idx += 128) {
        const int   n   = idx >> 6;
        const int   l0  = idx & 63;
        const float fl  = (float)l0;
        const float mag = expf(s_dre[n] * fl);
        const float th  = s_dim[n] * fl;
        smB[n     ][l0] = mag * cosf(th);
        smB[n + HN][l0] = mag * sinf(th);
    }

    // ---- Step 3: Aa[l1][k] = [2Wr | -2Wi], W = Ct * exp(dtA * 64 * l1) -----
    for (int idx = t; idx < HN * L1DIM; idx += 128) {
        const int   n   = idx >> 6;
        const int   l1  = idx & 63;
        const float s   = 64.0f * (float)l1;
        const float mag = expf(s_dre[n] * s);
        const float th  = s_dim[n] * s;
        const float Pr  = mag * cosf(th);
        const float Pi  = mag * sinf(th);
        const float Wr  = s_ctr[n] * Pr - s_cti[n] * Pi;
        const float Wi  = s_ctr[n] * Pi + s_cti[n] * Pr;
        smA[l1][n]      =  2.0f * Wr;
        smA[l1][n + HN] = -2.0f * Wi;
    }
    __syncthreads();

    // ---- Step 4: D(64x64) = Aa @ Bb via V_WMMA_F32_16X16X4_F32 -------------
    // Wave `wave` computes rows [16*wave, 16*wave+16) of D, all 64 cols.
    // A-frag layout (16x4 f32): lane 0-15 -> {K=0,K=1}, lane 16-31 -> {K=2,K=3},
    // row m = lane&15. B-frag mirrors with col n = lane&15.
    const int m_base = wave * 16;
    const int am     = lane & 15;
    const int kg     = (lane >> 4) * 2;    // 0 or 2

    v2f afrag[16];
#pragma unroll
    for (int ks = 0; ks < 16; ++ks) {
        afrag[ks].x = smA[m_base + am][ks * 4 + kg + 0];
        afrag[ks].y = smA[m_base + am][ks * 4 + kg + 1];
    }

    const int row_hi = (lane >> 4) * 8;    // C/D: VGPR v holds M = v + row_hi
#pragma unroll
    for (int ct = 0; ct < 4; ++ct) {
        const int nb = ct * 16 + (lane & 15);
        v8f acc = {0.f, 0.f, 0.f, 0.f, 0.f, 0.f, 0.f, 0.f};
#pragma unroll
        for (int ks = 0; ks < 16; ++ks) {
            v2f bfrag;
            bfrag.x = smB[ks * 4 + kg + 0][nb];
            bfrag.y = smB[ks * 4 + kg + 1][nb];
            acc = __builtin_amdgcn_wmma_f32_16x16x4_f32(
                /*neg_a=*/false, afrag[ks],
                /*neg_b=*/false, bfrag,
                /*c_mod=*/(short)0, acc,
                /*reuse_a=*/false, /*reuse_b=*/false);
        }
        // D[row][col] -> K_out[h*4096 + row*64 + col] (row = l1, col = l0)
        float* outp = K_out + (size_t)h * (L1DIM * LTILE)
                            + (size_t)(m_base + row_hi) * LTILE + nb;
#pragma unroll
        for (int v = 0; v < 8; ++v) {
            outp[v * LTILE] = acc[v];
        }
    }
}

extern "C" void kernel_launch(void* const* d_in, const int* in_sizes, int n_in,
                              void* d_out, int out_size, void* d_ws, size_t ws_size,
                              hipStream_t stream) {
    (void)n_in; (void)d_ws; (void)ws_size; (void)out_size;
    const float* C_real     = (const float*)d_in[0];
    const float* log_dt     = (const float*)d_in[1];
    const float* log_A_real = (const float*)d_in[2];
    const float* A_imag     = (const float*)d_in[3];
    // d_in[4] is L (=4096), baked into the 64x64 decomposition.
    float* K_out = (float*)d_out;

    const int H = in_sizes[1];             // log_dt has H elements (1024)
    s4d_vandermonde_wmma<<<dim3(H), dim3(128), 0, stream>>>(
        C_real, log_dt, log_A_real, A_imag, K_out);
}